// GaussianNoise_20229295964600
// MI455X (gfx1250) — compile-verified
//
#include <hip/hip_runtime.h>
#include <cstdint>

// out = x + sqrt(SIGMA) * N(0,1), SIGMA = 0.1
// B=16384, D=2048 -> 33,554,432 f32 elements (divisible by 4).
//
// Bandwidth-bound: 268 MB of traffic / 23.3 TB/s ~= 11.5 us floor on MI455X.
// Strategy: B128 non-temporal streaming loads/stores (stream > 192 MB L2, zero
// reuse), Philox4x32-10 counter RNG keyed on the float4 index (deterministic,
// launch-geometry independent), Box-Muller via fast TRANS32 ops that co-execute
// with the integer Philox VALU work.

typedef float f32x4 __attribute__((ext_vector_type(4)));

#define SQRT_SIGMA 0.31622776601683794f   // sqrt(0.1)
#define PHILOX_M0  0xD2511F53u
#define PHILOX_M1  0xCD9E8D57u
#define PHILOX_W0  0x9E3779B9u
#define PHILOX_W1  0xBB67AE85u

struct U4 { uint32_t a, b, c, d; };

__device__ __forceinline__ uint32_t mulhilo(uint32_t a, uint32_t b, uint32_t& hi) {
    uint64_t p = (uint64_t)a * (uint64_t)b;
    hi = (uint32_t)(p >> 32);
    return (uint32_t)p;
}

__device__ __forceinline__ U4 philox4x32_10(uint32_t c0, uint32_t c1,
                                            uint32_t c2, uint32_t c3,
                                            uint32_t k0, uint32_t k1) {
#pragma unroll
    for (int r = 0; r < 10; ++r) {
        uint32_t hi0, hi1;
        uint32_t lo0 = mulhilo(PHILOX_M0, c0, hi0);
        uint32_t lo1 = mulhilo(PHILOX_M1, c2, hi1);
        uint32_t n0 = hi1 ^ c1 ^ k0;
        uint32_t n1 = lo1;
        uint32_t n2 = hi0 ^ c3 ^ k1;
        uint32_t n3 = lo0;
        c0 = n0; c1 = n1; c2 = n2; c3 = n3;
        k0 += PHILOX_W0; k1 += PHILOX_W1;
    }
    return U4{c0, c1, c2, c3};
}

// Two independent standard normals from two u32s (Box-Muller).
__device__ __forceinline__ float2 box_muller(uint32_t a, uint32_t b) {
    const float TWO_NEG32 = 2.3283064365386963e-10f;  // 2^-32
    float u1 = ((float)a + 1.0f) * TWO_NEG32;         // (0, 1], log-safe
    float u2 = (float)b * TWO_NEG32;                  // [0, 1)
    float r  = __fsqrt_rn(-2.0f * __logf(u1));        // v_sqrt_f32 / v_log_f32
    float s, c;
    __sincosf(6.28318530717958647f * u2, &s, &c);     // v_sin_f32 / v_cos_f32
    return make_float2(r * c, r * s);
}

__global__ void __launch_bounds__(256)
gaussian_noise_kernel(const f32x4* __restrict__ x,
                      f32x4* __restrict__ out,
                      int nvec) {
    const int stride = (int)(gridDim.x * blockDim.x);
    for (int i = (int)(blockIdx.x * blockDim.x + threadIdx.x); i < nvec; i += stride) {
        // B128 non-temporal streaming load (TH=NT): no reuse, don't pollute L2.
        f32x4 v = __builtin_nontemporal_load(&x[i]);

        // One Philox4x32-10 block per float4: 4 u32 -> 4 normals.
        U4 r = philox4x32_10((uint32_t)i, 0x2eb1e0c5u, 0x9e3779b9u, 0xbb67ae85u,
                             42u, 0u);
        float2 z01 = box_muller(r.a, r.b);
        float2 z23 = box_muller(r.c, r.d);

        v.x += SQRT_SIGMA * z01.x;
        v.y += SQRT_SIGMA * z01.y;
        v.z += SQRT_SIGMA * z23.x;
        v.w += SQRT_SIGMA * z23.y;

        // B128 non-temporal streaming store.
        __builtin_nontemporal_store(v, &out[i]);
    }
}

extern "C" void kernel_launch(void* const* d_in, const int* in_sizes, int n_in,
                              void* d_out, int out_size, void* d_ws, size_t ws_size,
                              hipStream_t stream) {
    (void)in_sizes; (void)n_in; (void)d_ws; (void)ws_size;

    const f32x4* x   = (const f32x4*)d_in[0];
    f32x4*       out = (f32x4*)d_out;

    // out_size = 16384*2048 = 33,554,432; divisible by 4.
    const int nvec = out_size >> 2;

    const int block = 256;                       // 8 wave32 waves
    int grid = (nvec + block - 1) / block;       // 32768 blocks, 1 float4/thread
    if (grid > 65535 * 8) grid = 65535 * 8;      // grid-stride loop covers rest

    gaussian_noise_kernel<<<grid, block, 0, stream>>>(x, out, nvec);
}